// RefineDetMultiBoxLoss_35304631173495
// MI455X (gfx1250) — compile-verified
//
#include <hip/hip_runtime.h>
#include <math.h>

// ---------------------------------------------------------------------------
// RefineDet MultiBox loss for MI455X (gfx1250, wave32).
// Memory-bound pipeline:
//   - decode fused into every consumer (recompute beats 8.4 MB round-trip)
//   - single pass over the 44 MB odm_conf (logsumexp reused for mining + CE)
//   - per-batch hard-negative mining via exact 32-step bit-radix selection,
//     keys staged once in LDS (64 KB row << 320 KB/WGP)
//   - final reductions accumulated with v_wmma_f32_16x16x4_f32
// ---------------------------------------------------------------------------

typedef float v2f __attribute__((ext_vector_type(2)));
typedef float v8f __attribute__((ext_vector_type(8)));

#define VAR0 0.1f
#define VAR1 0.2f
#define THRESH 0.5f
#define THETA 0.01f

__device__ __forceinline__ void decode_box(const float* __restrict__ priors,
                                           const float* __restrict__ arm_loc,
                                           int b, int p, int P,
                                           float& d0, float& d1, float& d2, float& d3) {
  const float pr0 = priors[p * 4 + 0], pr1 = priors[p * 4 + 1];
  const float pr2 = priors[p * 4 + 2], pr3 = priors[p * 4 + 3];
  const size_t li = ((size_t)b * P + p) * 4;
  const float l0 = arm_loc[li + 0], l1 = arm_loc[li + 1];
  const float l2 = arm_loc[li + 2], l3 = arm_loc[li + 3];
  const float cx = pr0 + l0 * VAR0 * pr2;
  const float cy = pr1 + l1 * VAR0 * pr3;
  const float w = pr2 * expf(l2 * VAR1);
  const float h = pr3 * expf(l3 * VAR1);
  d0 = cx - w * 0.5f; d1 = cy - h * 0.5f;
  d2 = cx + w * 0.5f; d3 = cy + h * 0.5f;
}

__device__ __forceinline__ float iou_box(float t0, float t1, float t2, float t3,
                                         float d0, float d1, float d2, float d3) {
  const float ltx = fmaxf(t0, d0), lty = fmaxf(t1, d1);
  const float rbx = fminf(t2, d2), rby = fminf(t3, d3);
  const float wx = fmaxf(rbx - ltx, 0.0f), wy = fmaxf(rby - lty, 0.0f);
  const float inter = wx * wy;
  const float area_t = (t2 - t0) * (t3 - t1);
  const float area_d = (d2 - d0) * (d3 - d1);
  return inter / (area_t + area_d - inter);
}

// --- K1: per (b,g) argmax over P of IoU(truth_g, decoded prior) -------------
__global__ void k_best_prior(const float* __restrict__ arm_loc,
                             const float* __restrict__ priors,
                             const float* __restrict__ targets,
                             const int* __restrict__ num_gts,
                             int* __restrict__ bp_idx,
                             int B, int P, int G) {
  __shared__ float sv[256];
  __shared__ int si[256];
  const int b = blockIdx.x / G;
  const int g = blockIdx.x % G;
  const int ng = num_gts[b];
  if (g >= ng) {
    if (threadIdx.x == 0) bp_idx[b * G + g] = 0;
    return;
  }
  const size_t ti = ((size_t)b * G + g) * 5;
  const float t0 = targets[ti + 0], t1 = targets[ti + 1];
  const float t2 = targets[ti + 2], t3 = targets[ti + 3];
  float best = -1e30f;
  int bi = 0x7fffffff;
  for (int p = threadIdx.x; p < P; p += blockDim.x) {
    float d0, d1, d2, d3;
    decode_box(priors, arm_loc, b, p, P, d0, d1, d2, d3);
    const float ov = iou_box(t0, t1, t2, t3, d0, d1, d2, d3);
    if (ov > best || (ov == best && p < bi)) { best = ov; bi = p; }
  }
  sv[threadIdx.x] = best;
  si[threadIdx.x] = bi;
  __syncthreads();
  for (int s = blockDim.x / 2; s > 0; s >>= 1) {
    if (threadIdx.x < s) {
      const float v2 = sv[threadIdx.x + s];
      const int i2 = si[threadIdx.x + s];
      if (v2 > sv[threadIdx.x] || (v2 == sv[threadIdx.x] && i2 < si[threadIdx.x])) {
        sv[threadIdx.x] = v2; si[threadIdx.x] = i2;
      }
    }
    __syncthreads();
  }
  if (threadIdx.x == 0) bp_idx[b * G + g] = si[0];
}

// --- K2: per (b,p) max/argmax over G (first-max semantics) ------------------
__global__ void k_best_truth(const float* __restrict__ arm_loc,
                             const float* __restrict__ priors,
                             const float* __restrict__ targets,
                             const int* __restrict__ num_gts,
                             float* __restrict__ bt_ov,
                             int* __restrict__ bt_idx,
                             int B, int P, int G) {
  const int idx = blockIdx.x * blockDim.x + threadIdx.x;
  if (idx >= B * P) return;
  const int b = idx / P;
  const int p = idx - b * P;
  float d0, d1, d2, d3;
  decode_box(priors, arm_loc, b, p, P, d0, d1, d2, d3);
  const int ng = num_gts[b];
  float best = -3.0f;
  int bi = 0;
  for (int g = 0; g < G; ++g) {
    float val = -1.0f;
    if (g < ng) {
      const size_t ti = ((size_t)b * G + g) * 5;
      val = iou_box(targets[ti + 0], targets[ti + 1], targets[ti + 2], targets[ti + 3],
                    d0, d1, d2, d3);
    }
    if (val > best) { best = val; bi = g; }
  }
  bt_ov[idx] = best;
  bt_idx[idx] = bi;
}

// --- K3: force-match (serial per batch => exact "last write wins") ----------
__global__ void k_force(const int* __restrict__ bp_idx,
                        float* __restrict__ bt_ov,
                        int* __restrict__ bt_idx,
                        const int* __restrict__ num_gts,
                        int* __restrict__ num_pos,
                        int B, int P, int G) {
  const int b = blockIdx.x * blockDim.x + threadIdx.x;
  if (b >= B) return;
  num_pos[b] = 0;
  int ng = num_gts[b];
  if (ng > G) ng = G;
  for (int g = 0; g < ng; ++g) {
    const int p = bp_idx[b * G + g];
    bt_ov[b * P + p] = 2.0f;
    bt_idx[b * P + p] = g;
  }
}

// --- K4: encode, pos mask, smooth-L1 contrib, lc_raw (one odm_conf pass) ----
__global__ void k_match(const float* __restrict__ arm_loc,
                        const float* __restrict__ arm_conf,
                        const float* __restrict__ odm_loc,
                        const float* __restrict__ odm_conf,
                        const float* __restrict__ priors,
                        const float* __restrict__ targets,
                        const float* __restrict__ bt_ov,
                        const int* __restrict__ bt_idx,
                        int* __restrict__ posm,
                        float* __restrict__ lossl,
                        float* __restrict__ lc_raw,
                        int* __restrict__ num_pos,
                        int B, int P, int G, int C) {
  const int idx = blockIdx.x * blockDim.x + threadIdx.x;
  if (idx >= B * P) return;
  const int b = idx / P;
  const int p = idx - b * P;

  // decoded box -> center_size form
  float d0, d1, d2, d3;
  decode_box(priors, arm_loc, b, p, P, d0, d1, d2, d3);
  const float pcx = (d0 + d2) * 0.5f, pcy = (d1 + d3) * 0.5f;
  const float pcw = d2 - d0, pch = d3 - d1;

  const int g = bt_idx[idx];
  const float ov = bt_ov[idx];
  const size_t ti = ((size_t)b * G + g) * 5;
  const float m0 = targets[ti + 0], m1 = targets[ti + 1];
  const float m2 = targets[ti + 2], m3 = targets[ti + 3];
  const int lab = (int)targets[ti + 4];
  const int conf = (ov < THRESH) ? 0 : lab;

  // encode(matched, center_size(dec))
  const float e0 = ((m0 + m2) * 0.5f - pcx) / (VAR0 * pcw);
  const float e1 = ((m1 + m3) * 0.5f - pcy) / (VAR0 * pch);
  const float e2 = logf((m2 - m0) / pcw) / VAR1;
  const float e3 = logf((m3 - m1) / pch) / VAR1;

  // p_obj = softmax(arm_conf)[...,1]
  const float a0 = arm_conf[(size_t)idx * 2 + 0];
  const float a1 = arm_conf[(size_t)idx * 2 + 1];
  const float pobj = 1.0f / (1.0f + expf(a0 - a1));
  const bool pos = (conf > 0) && (pobj > THETA);
  posm[idx] = pos ? 1 : 0;

  // smooth-L1 contribution (0 when not pos, matching odm_loc*posf - loc_t*posf)
  float s = 0.0f;
  if (pos) {
    const size_t oi = (size_t)idx * 4;
    const float e[4] = {e0, e1, e2, e3};
    #pragma unroll
    for (int i = 0; i < 4; ++i) {
      const float d = fabsf(odm_loc[oi + i] - e[i]);
      s += (d < 1.0f) ? 0.5f * d * d : d - 0.5f;
    }
  }
  lossl[idx] = s;

  // lc_raw = logsumexp(row) - row[conf]   (single pass over odm_conf)
  const float* row = odm_conf + (size_t)idx * C;
  float mx = row[0];
  for (int c = 1; c < C; ++c) mx = fmaxf(mx, row[c]);
  float sum = 0.0f;
  for (int c = 0; c < C; ++c) sum += expf(row[c] - mx);
  lc_raw[idx] = (mx + logf(sum)) - row[conf];

  if (pos) atomicAdd(&num_pos[b], 1);
}

// --- K5: per-batch exact top-k selection, keys resident in LDS --------------
// Selection keys (pos ? 0 : lc_raw, clamped >= 0, reinterpreted as uint) for
// one batch row are 64 KB: staged once into dynamic LDS (320 KB/WGP on CDNA5),
// then the 33 counting passes + stable tie scan run entirely from LDS.
// Emits the fused CE contribution ce[p] = (pos|neg) ? lc_raw : 0 directly.
extern __shared__ unsigned char k_select_smem[];

__device__ int block_count_ge_lds(const unsigned* __restrict__ u_lds, int P,
                                  unsigned cand, int* scnt) {
  int c = 0;
  for (int p = threadIdx.x; p < P; p += blockDim.x)
    c += (u_lds[p] >= cand) ? 1 : 0;
  scnt[threadIdx.x] = c;
  __syncthreads();
  for (int s = blockDim.x / 2; s > 0; s >>= 1) {
    if (threadIdx.x < s) scnt[threadIdx.x] += scnt[threadIdx.x + s];
    __syncthreads();
  }
  const int tot = scnt[0];
  __syncthreads();
  return tot;
}

__global__ void k_select(const float* __restrict__ lc_raw,
                         const int* __restrict__ posm,
                         const int* __restrict__ num_pos,
                         float* __restrict__ ce,
                         int B, int P) {
  unsigned* u_lds = (unsigned*)k_select_smem;                       // P keys
  int* scnt = (int*)(k_select_smem + (size_t)P * sizeof(unsigned)); // 256 ints
  const int b = blockIdx.x;
  const int base = b * P;

  // stage keys
  for (int p = threadIdx.x; p < P; p += blockDim.x) {
    float v = posm[base + p] ? 0.0f : lc_raw[base + p];
    v = fmaxf(v, 0.0f);  // lc >= 0; keeps float bits order-isomorphic to uint
    u_lds[p] = __float_as_uint(v);
  }
  __syncthreads();

  int k = 3 * num_pos[b];
  if (k > P - 1) k = P - 1;
  if (k <= 0) {
    for (int p = threadIdx.x; p < P; p += blockDim.x)
      ce[base + p] = posm[base + p] ? lc_raw[base + p] : 0.0f;
    return;
  }

  // T = k-th largest key: max t with count(u >= t) >= k (32-step radix descent)
  unsigned t = 0u;
  for (int bit = 31; bit >= 0; --bit) {
    const unsigned cand = t | (1u << bit);
    const int c = block_count_ge_lds(u_lds, P, cand, scnt);
    if (c >= k) t = cand;
  }

  // strictly-greater keys are negatives; fuse mask into the CE contribution
  for (int p = threadIdx.x; p < P; p += blockDim.x) {
    const bool neg = u_lds[p] > t;
    const bool pos = posm[base + p] != 0;
    ce[base + p] = (pos || neg) ? lc_raw[base + p] : 0.0f;
  }
  __syncthreads();
  const int cntG = block_count_ge_lds(u_lds, P, t + 1u, scnt);
  if (threadIdx.x == 0) {
    int quota = k - cntG;  // ties at T admitted in stable (index) order
    for (int p = 0; p < P && quota > 0; ++p) {
      if (u_lds[p] == t) { ce[base + p] = lc_raw[base + p]; --quota; }
    }
  }
}

// --- K6: WMMA-accumulated reduction of both losses --------------------------
// Sum via D = A x Ones + C (v_wmma_f32_16x16x4_f32). With B = all-ones,
// sum over all 256 D entries = 16 * sum(A), so a full-wave shuffle reduce of
// the per-lane accumulators, times 1/16, is the exact chunk sum.
// N's tail (N % 64) is folded into block 0's fixed-order combine.
__global__ void k_reduce(const float* __restrict__ lossl,
                         const float* __restrict__ ce,
                         float* __restrict__ partial_l,
                         float* __restrict__ partial_c,
                         int N) {
  __shared__ float shl[8];
  __shared__ float shc[8];
  const int lane = threadIdx.x & 31;
  const int wib = threadIdx.x >> 5;
  const int wpb = blockDim.x >> 5;
  const int w = blockIdx.x * wpb + wib;
  const int W = gridDim.x * wpb;
  const int nchunk = N >> 6;  // full 64-element chunks

  v8f cl = {0.f, 0.f, 0.f, 0.f, 0.f, 0.f, 0.f, 0.f};
  v8f cc = {0.f, 0.f, 0.f, 0.f, 0.f, 0.f, 0.f, 0.f};
  v2f ones = {1.0f, 1.0f};

  for (int chunk = w; chunk < nchunk; chunk += W) {
    const int e = (chunk << 6) + (lane << 1);
    const v2f al = *reinterpret_cast<const v2f*>(lossl + e);
    const v2f ac = *reinterpret_cast<const v2f*>(ce + e);
    __builtin_prefetch(lossl + e + (W << 6), 0, 1);
    __builtin_prefetch(ce + e + (W << 6), 0, 1);
    cl = __builtin_amdgcn_wmma_f32_16x16x4_f32(false, al, false, ones, (short)0, cl,
                                               false, false);
    cc = __builtin_amdgcn_wmma_f32_16x16x4_f32(false, ac, false, ones, (short)0, cc,
                                               false, false);
  }
  float sl = 0.0f, sc = 0.0f;
  #pragma unroll
  for (int r = 0; r < 8; ++r) { sl += cl[r]; sc += cc[r]; }
  for (int off = 16; off > 0; off >>= 1) {
    sl += __shfl_xor(sl, off);
    sc += __shfl_xor(sc, off);
  }
  sl *= (1.0f / 16.0f);
  sc *= (1.0f / 16.0f);
  if (lane == 0) { shl[wib] = sl; shc[wib] = sc; }
  __syncthreads();
  if (threadIdx.x == 0) {
    float tl = 0.0f, tc = 0.0f;
    for (int i = 0; i < wpb; ++i) { tl += shl[i]; tc += shc[i]; }
    if (blockIdx.x == 0) {  // scalar tail (none when N % 64 == 0)
      for (int e = nchunk << 6; e < N; ++e) { tl += lossl[e]; tc += ce[e]; }
    }
    partial_l[blockIdx.x] = tl;
    partial_c[blockIdx.x] = tc;
  }
}

// --- K7: fixed-order final combine (deterministic) --------------------------
__global__ void k_final(const float* __restrict__ partial_l,
                        const float* __restrict__ partial_c,
                        const int* __restrict__ num_pos,
                        float* __restrict__ out, int nb, int B) {
  if (blockIdx.x == 0 && threadIdx.x == 0) {
    float sl = 0.0f, sc = 0.0f;
    for (int i = 0; i < nb; ++i) { sl += partial_l[i]; sc += partial_c[i]; }
    int np = 0;
    for (int b = 0; b < B; ++b) np += num_pos[b];
    const float Nf = (float)np;
    out[0] = sl / Nf;
    out[1] = sc / Nf;
  }
}

// ---------------------------------------------------------------------------
extern "C" void kernel_launch(void* const* d_in, const int* in_sizes, int n_in,
                              void* d_out, int out_size, void* d_ws, size_t ws_size,
                              hipStream_t stream) {
  const float* arm_loc  = (const float*)d_in[0];
  const float* arm_conf = (const float*)d_in[1];
  const float* odm_loc  = (const float*)d_in[2];
  const float* odm_conf = (const float*)d_in[3];
  const float* priors   = (const float*)d_in[4];
  const float* targets  = (const float*)d_in[5];
  const int*   num_gts  = (const int*)d_in[6];

  const int B = in_sizes[6];
  const int P = in_sizes[4] / 4;
  const int G = in_sizes[5] / (B * 5);
  const int C = in_sizes[3] / (B * P);
  const int NBP = B * P;
  const int RB = 64;  // reduction blocks

  char* w = (char*)d_ws;
  float* bt_ov     = (float*)w; w += sizeof(float) * NBP;
  int*   bt_idx    = (int*)w;   w += sizeof(int)   * NBP;
  int*   bp_idx    = (int*)w;   w += sizeof(int)   * B * G;
  int*   posm      = (int*)w;   w += sizeof(int)   * NBP;
  float* ce        = (float*)w; w += sizeof(float) * NBP;
  float* lc_raw    = (float*)w; w += sizeof(float) * NBP;
  float* lossl     = (float*)w; w += sizeof(float) * NBP;
  int*   num_pos   = (int*)w;   w += sizeof(int)   * B;
  float* partial_l = (float*)w; w += sizeof(float) * RB;
  float* partial_c = (float*)w; w += sizeof(float) * RB;

  const int nb = (NBP + 255) / 256;

  k_best_prior<<<B * G, 256, 0, stream>>>(arm_loc, priors, targets, num_gts,
                                          bp_idx, B, P, G);
  k_best_truth<<<nb, 256, 0, stream>>>(arm_loc, priors, targets, num_gts,
                                       bt_ov, bt_idx, B, P, G);
  k_force<<<1, 64, 0, stream>>>(bp_idx, bt_ov, bt_idx, num_gts, num_pos, B, P, G);
  k_match<<<nb, 256, 0, stream>>>(arm_loc, arm_conf, odm_loc, odm_conf, priors,
                                  targets, bt_ov, bt_idx, posm, lossl, lc_raw,
                                  num_pos, B, P, G, C);
  const size_t sel_smem = (size_t)P * sizeof(unsigned) + 256 * sizeof(int);
  k_select<<<B, 256, sel_smem, stream>>>(lc_raw, posm, num_pos, ce, B, P);
  k_reduce<<<RB, 256, 0, stream>>>(lossl, ce, partial_l, partial_c, NBP);
  k_final<<<1, 32, 0, stream>>>(partial_l, partial_c, num_pos, (float*)d_out, RB, B);
}